// ResidualGATLayer_88527865905436
// MI455X (gfx1250) — compile-verified
//
#include <hip/hip_runtime.h>
#include <hip/hip_bf16.h>

// ---------------------------------------------------------------------------
// ResidualGATLayer for MI455X (gfx1250, wave32).
//   h = x @ W                       -> V_WMMA_F32_16X16X4_F32 (exact fp32)
//   alpha_s/d = <h, a_src/dst>      -> wave-per-node shuffle reduction
//   segment softmax over dst        -> ordered-uint atomicMax + atomicAdd
//   out = elu(scatter(alpha*h[src]) + bias) + x
// ---------------------------------------------------------------------------

typedef __attribute__((ext_vector_type(2))) float v2f;
typedef __attribute__((ext_vector_type(8))) float v8f;

#define IN_CH 128   // = HEADS * OUT_CH, needed for residual
#define HEADS 4
#define OUTC  32

// ---- monotonic float <-> uint mapping so unsigned atomicMax == float max ----
__device__ __forceinline__ unsigned enc_f32(float f) {
    unsigned b = __float_as_uint(f);
    return (b & 0x80000000u) ? ~b : (b | 0x80000000u);
}
__device__ __forceinline__ float dec_f32(unsigned u) {
    return (u & 0x80000000u) ? __uint_as_float(u & 0x7fffffffu)
                             : __uint_as_float(~u);
}

// ---------------------------------------------------------------------------
// Kernel 1: h = x @ W via V_WMMA_F32_16X16X4_F32.
// Block = 256 threads = 8 waves. W (128x128 f32 = 64KB) is staged in LDS,
// pre-swizzled so each 4x16 B-fragment is one contiguous ds_load_b64 per lane.
// Each wave computes a 16-row strip across all 128 output columns
// (8 accumulators of v8f), K-loop of 32 steps -> 256 WMMAs per wave.
// ---------------------------------------------------------------------------
__global__ __launch_bounds__(256) void gat_gemm_wmma(
    const float* __restrict__ x, const float* __restrict__ W,
    float* __restrict__ h, int N)
{
    __shared__ float wlds[IN_CH * IN_CH];   // 64 KB (WGP has 320 KB)

    const int tid = threadIdx.x;

    // Cooperative load of W with fragment swizzle:
    // element (k, n) -> wlds[((k/4)*8 + n/16)*64 + lane*2 + v],
    //   half = (k%4)/2, v = (k%4)%2, lane = half*16 + (n%16).
    for (int i = tid * 4; i < IN_CH * IN_CH; i += 256 * 4) {
        const float4 w4 = *(const float4*)(W + i);
        const int k = i >> 7;
        const int n = i & 127;
        const int ktile = k >> 2;
        const int kk = k & 3;
        const int half = kk >> 1, v = kk & 1;
        const float wv[4] = {w4.x, w4.y, w4.z, w4.w};
#pragma unroll
        for (int j = 0; j < 4; ++j) {
            const int nn = n + j;
            const int lane = half * 16 + (nn & 15);
            wlds[((ktile * 8) + (nn >> 4)) * 64 + lane * 2 + v] = wv[j];
        }
    }
    __syncthreads();

    const int wave = tid >> 5;
    const int lane = tid & 31;
    const int half = lane >> 4;     // K-half selector for A/B fragments
    const int mlan = lane & 15;

    const long rowTile = (long)blockIdx.x * 8 + wave;
    if (rowTile * 16 >= N) return;  // after the only barrier -> safe

    v8f acc[8] = {};                // 16 rows x 128 cols of f32 accumulation

    // A-fragment rows: lanes 0-15 and 16-31 both index M = 0..15.
    const long arow = min(rowTile * 16 + mlan, (long)N - 1);
    const float* __restrict__ xrow = x + arow * IN_CH;

    for (int k0 = 0; k0 < IN_CH; k0 += 4) {
        // A 16x4 f32 fragment: lane<16 holds K=k0+{0,1}; lane>=16 K=k0+{2,3}.
        const v2f a = *(const v2f*)(xrow + k0 + half * 2);
        const float* __restrict__ bb = wlds + (k0 >> 2) * (8 * 64) + lane * 2;
#pragma unroll
        for (int nt = 0; nt < 8; ++nt) {
            const v2f b = *(const v2f*)(bb + nt * 64);   // ds_load_b64
            acc[nt] = __builtin_amdgcn_wmma_f32_16x16x4_f32(
                false, a, false, b, (short)0, acc[nt], false, false);
        }
    }

    // C/D layout: VGPR r -> M = r (lanes 0-15) / r+8 (lanes 16-31), N = mlan.
#pragma unroll
    for (int nt = 0; nt < 8; ++nt) {
#pragma unroll
        for (int r = 0; r < 8; ++r) {
            const long row = rowTile * 16 + r + half * 8;
            if (row < N) h[row * IN_CH + nt * 16 + mlan] = acc[nt][r];
        }
    }
}

// ---------------------------------------------------------------------------
// Kernel 2: alpha_s[n,hd] = <h[n,hd,:], a_src[hd,:]>, same for a_dst.
// One wave per node; lane l owns channels 4l..4l+3 (flat == a-vector index),
// head = l/8; 8-lane shuffle reduction.
// ---------------------------------------------------------------------------
__global__ __launch_bounds__(256) void gat_alpha(
    const float* __restrict__ h, const float* __restrict__ a_src,
    const float* __restrict__ a_dst, float* __restrict__ as,
    float* __restrict__ ad, int N)
{
    const long node = (long)blockIdx.x * 8 + (threadIdx.x >> 5);
    if (node >= N) return;
    const int lane = threadIdx.x & 31;

    const float4 hv = *(const float4*)(h + node * IN_CH + lane * 4);
    const float4 s4 = *(const float4*)(a_src + lane * 4);
    const float4 d4 = *(const float4*)(a_dst + lane * 4);
    float ps = hv.x * s4.x + hv.y * s4.y + hv.z * s4.z + hv.w * s4.w;
    float pd = hv.x * d4.x + hv.y * d4.y + hv.z * d4.z + hv.w * d4.w;
#pragma unroll
    for (int off = 1; off < 8; off <<= 1) {
        ps += __shfl_xor(ps, off, 32);
        pd += __shfl_xor(pd, off, 32);
    }
    if ((lane & 7) == 0) {
        const int hd = lane >> 3;
        as[node * HEADS + hd] = ps;
        ad[node * HEADS + hd] = pd;
    }
}

// ---------------------------------------------------------------------------
// Kernel 3: zero the accumulator (d_out) and softmax state. enc(-inf) > 0,
// so 0 is a safe identity for the unsigned max encoding.
// ---------------------------------------------------------------------------
__global__ void gat_init(float* __restrict__ out, unsigned* __restrict__ m,
                         float* __restrict__ denom, int N)
{
    const long stride = (long)gridDim.x * blockDim.x;
    const long i0 = (long)blockIdx.x * blockDim.x + threadIdx.x;
    for (long j = i0; j < (long)N * IN_CH; j += stride) out[j] = 0.0f;
    for (long j = i0; j < (long)N * HEADS; j += stride) {
        m[j] = 0u;
        denom[j] = 0.0f;
    }
}

__device__ __forceinline__ void edge_endpoints(const long long* __restrict__ ei,
                                               long e, long NE,
                                               long& src, long& dst)
{
    if (e < NE) { src = (long)ei[e]; dst = (long)ei[NE + e]; }
    else        { src = dst = e - NE; }        // self-loops appended
}

// ---------------------------------------------------------------------------
// Kernel 4: segment max of LeakyReLU(alpha_s[src] + alpha_d[dst]) over dst.
// ---------------------------------------------------------------------------
__global__ void gat_segmax(const long long* __restrict__ ei,
                           const float* __restrict__ as,
                           const float* __restrict__ ad,
                           unsigned* __restrict__ m, long NE, long ETOT)
{
    const long i = (long)blockIdx.x * blockDim.x + threadIdx.x;
    if (i >= ETOT * HEADS) return;
    const long e = i >> 2;
    const int hd = (int)(i & 3);
    long src, dst;
    edge_endpoints(ei, e, NE, src, dst);
    float v = as[src * HEADS + hd] + ad[dst * HEADS + hd];
    v = v > 0.0f ? v : 0.2f * v;               // LeakyReLU(0.2)
    atomicMax(m + dst * HEADS + hd, enc_f32(v));
}

// ---------------------------------------------------------------------------
// Kernel 5: denom[dst,hd] += exp(e - m[dst,hd])
// ---------------------------------------------------------------------------
__global__ void gat_segsum(const long long* __restrict__ ei,
                           const float* __restrict__ as,
                           const float* __restrict__ ad,
                           const unsigned* __restrict__ m,
                           float* __restrict__ denom, long NE, long ETOT)
{
    const long i = (long)blockIdx.x * blockDim.x + threadIdx.x;
    if (i >= ETOT * HEADS) return;
    const long e = i >> 2;
    const int hd = (int)(i & 3);
    long src, dst;
    edge_endpoints(ei, e, NE, src, dst);
    float v = as[src * HEADS + hd] + ad[dst * HEADS + hd];
    v = v > 0.0f ? v : 0.2f * v;
    const float p = expf(v - dec_f32(m[dst * HEADS + hd]));
    atomicAdd(denom + dst * HEADS + hd, p);
}

// ---------------------------------------------------------------------------
// Kernel 6: out[dst,:] += alpha * h[src,:]. One wave per edge; lane l owns
// channels 4l..4l+3 (head = l/8): float4 gather of h[src] + 4 fp32 atomics.
// This is the bandwidth-dominant pass (~435 MB gather, accumulator in L2).
// ---------------------------------------------------------------------------
__global__ __launch_bounds__(256) void gat_scatter(
    const long long* __restrict__ ei, const float* __restrict__ as,
    const float* __restrict__ ad, const unsigned* __restrict__ m,
    const float* __restrict__ denom, const float* __restrict__ h,
    float* __restrict__ out, long NE, long ETOT)
{
    const long e = (long)blockIdx.x * 8 + (threadIdx.x >> 5);
    if (e >= ETOT) return;
    const int lane = threadIdx.x & 31;
    const int hd = lane >> 3;

    long src, dst;
    edge_endpoints(ei, e, NE, src, dst);

    float v = as[src * HEADS + hd] + ad[dst * HEADS + hd];
    v = v > 0.0f ? v : 0.2f * v;
    const float p = expf(v - dec_f32(m[dst * HEADS + hd]));
    const float alpha = p / (denom[dst * HEADS + hd] + 1e-16f);

    const float4 hv = *(const float4*)(h + src * IN_CH + lane * 4);
    float* o = out + dst * IN_CH + lane * 4;
    atomicAdd(o + 0, alpha * hv.x);
    atomicAdd(o + 1, alpha * hv.y);
    atomicAdd(o + 2, alpha * hv.z);
    atomicAdd(o + 3, alpha * hv.w);
}

// ---------------------------------------------------------------------------
// Kernel 7: out = elu(out + bias) + x  (in place over the accumulator)
// ---------------------------------------------------------------------------
__global__ void gat_finalize(const float* __restrict__ x,
                             const float* __restrict__ bias,
                             float* __restrict__ out, int N)
{
    const long stride = (long)gridDim.x * blockDim.x;
    for (long j = (long)blockIdx.x * blockDim.x + threadIdx.x;
         j < (long)N * IN_CH; j += stride) {
        float v = out[j] + bias[j & (IN_CH - 1)];
        v = v > 0.0f ? v : expm1f(v);          // ELU(alpha=1)
        out[j] = v + x[j];
    }
}

// ---------------------------------------------------------------------------
extern "C" void kernel_launch(void* const* d_in, const int* in_sizes, int n_in,
                              void* d_out, int out_size, void* d_ws, size_t ws_size,
                              hipStream_t stream)
{
    const float*     x      = (const float*)d_in[0];
    const long long* ei     = (const long long*)d_in[1];   // int64 [2, E]
    const float*     W      = (const float*)d_in[2];
    const float*     a_src  = (const float*)d_in[3];
    const float*     a_dst  = (const float*)d_in[4];
    const float*     bias   = (const float*)d_in[5];

    const int  N    = in_sizes[0] / IN_CH;
    const long NE   = in_sizes[1] / 2;
    const long ETOT = NE + N;                  // + self-loops

    // workspace layout (floats): h | alpha_s | alpha_d | denom | m(uint)
    float*    h     = (float*)d_ws;
    float*    as    = h  + (long)N * IN_CH;
    float*    ad    = as + (long)N * HEADS;
    float*    denom = ad + (long)N * HEADS;
    unsigned* m     = (unsigned*)(denom + (long)N * HEADS);

    float* out = (float*)d_out;

    // 1) GEMM: 8 row-tiles (of 16 rows) per block
    {
        const int rowTiles = (N + 15) / 16;
        const int blocks = (rowTiles + 7) / 8;
        gat_gemm_wmma<<<blocks, 256, 0, stream>>>(x, W, h, N);
    }
    // 2) attention dot products (wave per node)
    gat_alpha<<<(N + 7) / 8, 256, 0, stream>>>(h, a_src, a_dst, as, ad, N);
    // 3) clear accumulator + softmax state
    gat_init<<<2048, 256, 0, stream>>>(out, m, denom, N);
    // 4) segment max
    {
        const long t = ETOT * HEADS;
        gat_segmax<<<(unsigned)((t + 255) / 256), 256, 0, stream>>>(ei, as, ad, m, NE, ETOT);
    }
    // 5) segment sum of exp
    {
        const long t = ETOT * HEADS;
        gat_segsum<<<(unsigned)((t + 255) / 256), 256, 0, stream>>>(ei, as, ad, m, denom, NE, ETOT);
    }
    // 6) weighted message scatter (wave per edge)
    gat_scatter<<<(unsigned)((ETOT + 7) / 8), 256, 0, stream>>>(
        ei, as, ad, m, denom, h, out, NE, ETOT);
    // 7) bias + ELU + residual
    gat_finalize<<<2048, 256, 0, stream>>>(x, bias, out, N);
}